// MultiLayerGCN_48773648613817
// MI455X (gfx1250) — compile-verified
//
#include <hip/hip_runtime.h>

typedef __attribute__((ext_vector_type(2))) float v2f;
typedef __attribute__((ext_vector_type(8))) float v8f;

#define GDIM 128  // feature dimension (reference DIM = 128)

// ---------------------------------------------------------------------------
// Degree / normalization helpers
// ---------------------------------------------------------------------------
__global__ void gcn_init_deg(float* __restrict__ deg, int n) {
  int i = blockIdx.x * blockDim.x + threadIdx.x;
  if (i < n) deg[i] = 1.0f;  // self-loop contributes 1
}

__global__ void gcn_deg_accum(float* __restrict__ deg, const int* __restrict__ col, int e) {
  int i = blockIdx.x * blockDim.x + threadIdx.x;
  if (i < e) atomicAdd(&deg[col[i]], 1.0f);  // exact for deg < 2^24
}

__global__ void gcn_deg_to_dis(float* __restrict__ deg, int n) {
  int i = blockIdx.x * blockDim.x + threadIdx.x;
  if (i < n) deg[i] = rsqrtf(deg[i]);  // deg >= 1 guaranteed
}

__global__ void gcn_zero(float* __restrict__ p, long long n) {
  long long i = (long long)blockIdx.x * blockDim.x + threadIdx.x;
  if (i < n) p[i] = 0.0f;
}

// ---------------------------------------------------------------------------
// H = X @ W   (nrows x 128) @ (128 x 128), fp32 WMMA 16x16x4.
// One wave per 16-row strip; 8 accumulator tiles cover all 128 output cols.
// A layout (16x4 f32): lane L (L<16): M=L, VGPR0->K=2*half, VGPR1->K=2*half+1.
// B/C layout: row striped across lanes, N = lane%16.
// A-row index is CLAMPED (not predicated) so the k-loop has no exec masking;
// duplicate rows from clamping are never stored (store-side guard).
// ---------------------------------------------------------------------------
__global__ __launch_bounds__(256) void gcn_gemm_wmma(const float* __restrict__ X,
                                                     const float* __restrict__ Wm,
                                                     float* __restrict__ H,
                                                     int nrows) {
  const int lane = threadIdx.x & 31;
  const int wave = threadIdx.x >> 5;
  const int nstrips = (nrows + 15) >> 4;
  const int strip = blockIdx.x * 8 + wave;
  if (strip >= nstrips) return;

  const int lm = lane & 15;
  const int half = lane >> 4;

  int arow = strip * 16 + lm;
  if (arow > nrows - 1) arow = nrows - 1;  // clamp: always-in-bounds A loads
  const float* xrow = X + (long long)arow * GDIM;

  v8f acc[8] = {};  // 8 x (16x16 f32) accumulators = full 16x128 strip

#pragma unroll 4
  for (int k0 = 0; k0 < GDIM; k0 += 4) {
    const int ka = k0 + 2 * half;  // this lane-half's K pair
    const v2f a = *(const v2f*)(xrow + ka);
    const float* w0 = Wm + (long long)ka * GDIM + lm;  // row K=ka
    const float* w1 = w0 + GDIM;                       // row K=ka+1
#pragma unroll
    for (int t = 0; t < 8; ++t) {
      v2f bv;
      bv.x = w0[t * 16];
      bv.y = w1[t * 16];
      acc[t] = __builtin_amdgcn_wmma_f32_16x16x4_f32(
          /*neg_a=*/false, a, /*neg_b=*/false, bv,
          /*c_mod=*/(short)0, acc[t], /*reuse_a=*/false, /*reuse_b=*/false);
    }
  }

  // C/D layout: VGPR r -> M = strip*16 + r + 8*half, N = t*16 + lm
#pragma unroll
  for (int t = 0; t < 8; ++t) {
#pragma unroll
    for (int r = 0; r < 8; ++r) {
      const int m = strip * 16 + r + 8 * half;
      if (m < nrows) H[(long long)m * GDIM + t * 16 + lm] = acc[t][r];
    }
  }
}

// ---------------------------------------------------------------------------
// Edge scatter: one wave per edge, float4 per lane (128 floats per edge).
// agg[col] += h[row] * dis[row]*dis[col]  via global_atomic_add_f32 (L2).
// ---------------------------------------------------------------------------
__global__ __launch_bounds__(256) void gcn_scatter(const float4* __restrict__ h4,
                                                   const float* __restrict__ dis,
                                                   const int* __restrict__ row,
                                                   const int* __restrict__ col,
                                                   float* __restrict__ agg, int e) {
  const int gid = blockIdx.x * blockDim.x + threadIdx.x;
  const int edge = gid >> 5;
  const int lane = threadIdx.x & 31;
  if (edge >= e) return;

  const int r = row[edge];
  const int c = col[edge];
  const float norm = dis[r] * dis[c];

  const float4 v = h4[(long long)r * (GDIM / 4) + lane];
  float* dst = agg + (long long)c * GDIM + lane * 4;
  atomicAdd(dst + 0, v.x * norm);
  atomicAdd(dst + 1, v.y * norm);
  atomicAdd(dst + 2, v.z * norm);
  atomicAdd(dst + 3, v.w * norm);
}

// ---------------------------------------------------------------------------
// out = relu(agg + h * dis^2 (self-loop) + bias)
// ---------------------------------------------------------------------------
__global__ void gcn_finalize(const float* __restrict__ agg, const float* __restrict__ h,
                             const float* __restrict__ dis, const float* __restrict__ bias,
                             float* __restrict__ out, int n) {
  const long long idx = (long long)blockIdx.x * blockDim.x + threadIdx.x;
  const long long total = (long long)n * GDIM;
  if (idx >= total) return;
  const int i = (int)(idx >> 7);   // /128
  const int d = (int)(idx & 127);  // %128
  const float s = dis[i];
  const float v = agg[idx] + h[idx] * s * s + bias[d];
  out[idx] = v > 0.0f ? v : 0.0f;
}

// ---------------------------------------------------------------------------
// Host-side orchestration (graph-capture safe: only kernel launches on stream)
// ---------------------------------------------------------------------------
extern "C" void kernel_launch(void* const* d_in, const int* in_sizes, int n_in,
                              void* d_out, int out_size, void* d_ws, size_t ws_size,
                              hipStream_t stream) {
  const float* x0 = (const float*)d_in[0];   // (N, 128) f32
  const int* ei = (const int*)d_in[1];       // (L, 2, E) int32
  const float* W = (const float*)d_in[2];    // (L, 128, 128) f32
  const float* b = (const float*)d_in[3];    // (L, 128) f32

  const int N = in_sizes[0] / GDIM;
  const int L = in_sizes[2] / (GDIM * GDIM);
  const int E = in_sizes[1] / (2 * L);

  float* out = (float*)d_out;
  // Workspace layout: dis[N] | h[N*128] | agg[N*128]
  float* dis = (float*)d_ws;
  float* h = dis + N;
  float* agg = h + (long long)N * GDIM;

  const int nstrips = (N + 15) >> 4;
  const int gemm_blocks = (nstrips + 7) / 8;
  const int n_blocks = (N + 255) / 256;
  const int e_blocks = (E + 255) / 256;
  const long long nd = (long long)N * GDIM;
  const int nd_blocks = (int)((nd + 255) / 256);
  const int scat_blocks = (int)(((long long)E * 32 + 255) / 256);

  for (int l = 0; l < L; ++l) {
    const float* xin = (l == 0) ? x0 : out;          // d_out doubles as layer input
    const int* row = ei + (long long)l * 2 * E;      // edge_index[l][0]: sources
    const int* col = row + E;                        // edge_index[l][1]: targets
    const float* Wl = W + (long long)l * GDIM * GDIM;
    const float* bl = b + (long long)l * GDIM;

    gcn_init_deg<<<n_blocks, 256, 0, stream>>>(dis, N);
    gcn_deg_accum<<<e_blocks, 256, 0, stream>>>(dis, col, E);
    gcn_deg_to_dis<<<n_blocks, 256, 0, stream>>>(dis, N);

    gcn_gemm_wmma<<<gemm_blocks, 256, 0, stream>>>(xin, Wl, h, N);

    gcn_zero<<<nd_blocks, 256, 0, stream>>>(agg, nd);
    gcn_scatter<<<scat_blocks, 256, 0, stream>>>((const float4*)h, dis, row, col, agg, E);

    gcn_finalize<<<nd_blocks, 256, 0, stream>>>(agg, h, dis, bl, out, N);
  }
}